// GConv_71794673320215
// MI455X (gfx1250) — compile-verified
//
#include <hip/hip_runtime.h>

#define H 128
#define LAYERS 3
#define GRAPHS 128
#define BN_EPS 1e-5f

typedef __attribute__((ext_vector_type(2))) float v2f;
typedef __attribute__((ext_vector_type(8))) float v8f;

// ---------------------------------------------------------------------------
// Generic zero-fill
// ---------------------------------------------------------------------------
__global__ void zero_kernel(float* p, long long n) {
    long long i = (long long)blockIdx.x * blockDim.x + threadIdx.x;
    if (i < n) p[i] = 0.0f;
}

__global__ void fill_one_kernel(float* p, int n) {
    int i = blockIdx.x * blockDim.x + threadIdx.x;
    if (i < n) p[i] = 1.0f;   // degree starts at 1.0 (self-loop weight)
}

// ---------------------------------------------------------------------------
// Degree / normalization precompute
// ---------------------------------------------------------------------------
__global__ void deg_accum_kernel(const int* __restrict__ col,
                                 const float* __restrict__ ea,
                                 float* __restrict__ deg, int E) {
    int e = blockIdx.x * blockDim.x + threadIdx.x;
    if (e < E) atomicAdd(&deg[col[e]], ea[e]);
}

__global__ void deg_finalize_kernel(float* __restrict__ dinv,
                                    float* __restrict__ selfn, int n) {
    int i = blockIdx.x * blockDim.x + threadIdx.x;
    if (i < n) {
        float d = dinv[i];           // holds degree at this point
        dinv[i]  = rsqrtf(d);
        selfn[i] = 1.0f / d;
    }
}

__global__ void edge_norm_kernel(const int* __restrict__ row,
                                 const int* __restrict__ col,
                                 const float* __restrict__ ea,
                                 const float* __restrict__ dinv,
                                 float* __restrict__ normw, int E) {
    int e = blockIdx.x * blockDim.x + threadIdx.x;
    if (e < E) normw[e] = dinv[row[e]] * ea[e] * dinv[col[e]];
}

// ---------------------------------------------------------------------------
// WMMA f32 GEMM:  D[m, dcol0 + n] = act( A[m,:] @ Wm[:,n] + bias[n] )
// A: [nrows x 128] ld=128, Wm: [128 x 128], D row stride = ldd.
// Block = 256 threads (8 wave32). Wave w computes the 16x16 tile at columns
// w*16. A 16-row tile is staged in LDS (padded to 129 floats/row to dodge
// bank conflicts). K-loop = 32 x v_wmma_f32_16x16x4_f32.
// ---------------------------------------------------------------------------
__global__ __launch_bounds__(256)
void gemm_wmma_kernel(const float* __restrict__ A,
                      const float* __restrict__ Wm,
                      const float* __restrict__ bias,
                      float* __restrict__ D,
                      int nrows, int ldd, int dcol0, int do_relu) {
    __shared__ float As[16 * 129];

    const int m0  = blockIdx.x * 16;
    const int tid = threadIdx.x;

    // Cooperative load of the 16x128 A tile (zero-pad tail rows)
    for (int i = tid; i < 16 * 128; i += 256) {
        int r = i >> 7, c = i & 127;
        int gr = m0 + r;
        As[r * 129 + c] = (gr < nrows) ? A[(size_t)gr * H + c] : 0.0f;
    }
    __syncthreads();

    const int wave = tid >> 5;        // 0..7 -> column tile
    const int lane = tid & 31;
    const int half = lane >> 4;       // 0: K pair {0,1}, 1: K pair {2,3}
    const int l16  = lane & 15;
    const int n0   = wave * 16;

    v8f acc = {};
    #pragma unroll
    for (int k = 0; k < H; k += 4) {
        v2f a, b;
        // A 16x4: lanes 0-15 row M=l16 K={k,k+1}; lanes 16-31 K={k+2,k+3}
        a.x = As[l16 * 129 + k + 2 * half + 0];
        a.y = As[l16 * 129 + k + 2 * half + 1];
        // B 4x16: row-striped across lanes, same half split over K
        b.x = Wm[(size_t)(k + 2 * half + 0) * H + n0 + l16];
        b.y = Wm[(size_t)(k + 2 * half + 1) * H + n0 + l16];
        acc = __builtin_amdgcn_wmma_f32_16x16x4_f32(
                  false, a, false, b, (short)0, acc, false, false);
    }

    // C/D layout: VGPR v -> row M = half*8 + v, column N = n0 + l16
    float bv = bias ? bias[n0 + l16] : 0.0f;
    #pragma unroll
    for (int v = 0; v < 8; ++v) {
        int m = m0 + half * 8 + v;
        if (m < nrows) {
            float val = acc[v] + bv;
            if (do_relu) val = fmaxf(val, 0.0f);
            D[(size_t)m * ldd + dcol0 + n0 + l16] = val;
        }
    }
}

// ---------------------------------------------------------------------------
// agg[n][c] = selfnorm[n] * h[n][c] + bias[c]
// ---------------------------------------------------------------------------
__global__ void agg_init_kernel(const float* __restrict__ h,
                                const float* __restrict__ selfn,
                                const float* __restrict__ bias,
                                float* __restrict__ agg, int n) {
    long long idx = (long long)blockIdx.x * blockDim.x + threadIdx.x;
    if (idx < (long long)n * H) {
        int node = (int)(idx >> 7);
        int c    = (int)(idx & (H - 1));
        agg[idx] = selfn[node] * h[idx] + bias[c];
    }
}

// ---------------------------------------------------------------------------
// Edge scatter: agg[col[e]] += norm[e] * h[row[e]]   (32 threads / edge,
// float4 payload each). L2 (192MB) holds agg (51MB) so atomics stay on-chip.
// ---------------------------------------------------------------------------
__global__ void scatter_kernel(const int* __restrict__ row,
                               const int* __restrict__ col,
                               const float* __restrict__ normw,
                               const float* __restrict__ h,
                               float* __restrict__ agg, int E) {
    long long t = (long long)blockIdx.x * blockDim.x + threadIdx.x;
    int e = (int)(t >> 5);
    if (e >= E) return;
    int fq = ((int)t & 31) << 2;
    int r = row[e], c = col[e];
    float w = normw[e];
    float4 hv = *(const float4*)(h + (size_t)r * H + fq);
    float* dst = agg + (size_t)c * H + fq;
    atomicAdd(dst + 0, w * hv.x);
    atomicAdd(dst + 1, w * hv.y);
    atomicAdd(dst + 2, w * hv.z);
    atomicAdd(dst + 3, w * hv.w);
}

// ---------------------------------------------------------------------------
// In-place ReLU + batchnorm statistics. Each block owns 64 rows; two threads
// per column accumulate privately, LDS-combine, then ONE atomic per column.
// stat[0..127] = sum, stat[128..255] = sum of squares.
// ---------------------------------------------------------------------------
__global__ __launch_bounds__(256)
void relu_stats_kernel(float* __restrict__ z, float* __restrict__ stat, int n) {
    __shared__ float s1[256];
    __shared__ float s2[256];
    int c   = threadIdx.x & (H - 1);
    int sub = threadIdx.x >> 7;          // 0 or 1
    int r0  = blockIdx.x * 64;
    int rend = r0 + 64; if (rend > n) rend = n;

    float sum = 0.0f, sq = 0.0f;
    for (int r = r0 + sub; r < rend; r += 2) {
        size_t idx = (size_t)r * H + c;
        float v = fmaxf(z[idx], 0.0f);
        z[idx] = v;
        sum += v;
        sq  += v * v;
    }
    s1[threadIdx.x] = sum;
    s2[threadIdx.x] = sq;
    __syncthreads();
    if (sub == 0) {
        atomicAdd(&stat[c],       s1[c] + s1[c + 128]);
        atomicAdd(&stat[H + c],   s2[c] + s2[c + 128]);
    }
}

// stat -> (scale, shift) in place
__global__ void bn_finalize_kernel(float* __restrict__ stat,
                                   const float* __restrict__ gamma,
                                   const float* __restrict__ beta, int n) {
    int c = threadIdx.x;
    if (c < H) {
        float inv_n = 1.0f / (float)n;
        float mean = stat[c] * inv_n;
        float var  = stat[H + c] * inv_n - mean * mean;
        float rstd = rsqrtf(var + BN_EPS);
        float scale = rstd * gamma[c];
        stat[c]     = scale;
        stat[H + c] = beta[c] - mean * scale;
    }
}

// z_next = z*scale + shift; also write z_cat output slice (stride 512)
__global__ void bn_apply_kernel(const float* __restrict__ zin,
                                const float* __restrict__ stat,
                                float* __restrict__ zout,
                                float* __restrict__ zcat,
                                int layer, int n) {
    long long idx = (long long)blockIdx.x * blockDim.x + threadIdx.x;
    if (idx < (long long)n * H) {
        int node = (int)(idx >> 7);
        int c    = (int)(idx & (H - 1));
        float v = zin[idx] * stat[c] + stat[H + c];
        zout[idx] = v;
        zcat[(size_t)node * (H * (LAYERS + 1)) + layer * H + c] = v;
    }
}

// ---------------------------------------------------------------------------
// Graph pooling: g_cat[batch[n]] += z_cat[n]  (float4 per thread)
// ---------------------------------------------------------------------------
__global__ void pool_kernel(const float* __restrict__ zcat,
                            const int* __restrict__ batch,
                            float* __restrict__ gcat, int n) {
    const int C = H * (LAYERS + 1);           // 512
    long long t = (long long)blockIdx.x * blockDim.x + threadIdx.x;
    int node = (int)(t >> 7);
    if (node >= n) return;
    int c4 = ((int)t & 127) << 2;
    int g = batch[node];
    float4 v = *(const float4*)(zcat + (size_t)node * C + c4);
    float* dst = gcat + (size_t)g * C + c4;
    atomicAdd(dst + 0, v.x);
    atomicAdd(dst + 1, v.y);
    atomicAdd(dst + 2, v.z);
    atomicAdd(dst + 3, v.w);
}

// ---------------------------------------------------------------------------
// Host-side orchestration
// ---------------------------------------------------------------------------
extern "C" void kernel_launch(void* const* d_in, const int* in_sizes, int n_in,
                              void* d_out, int out_size, void* d_ws, size_t ws_size,
                              hipStream_t stream) {
    const float* x        = (const float*)d_in[0];
    const int*   eidx     = (const int*)  d_in[1];
    const float* eattr    = (const float*)d_in[2];
    const int*   batch    = (const int*)  d_in[3];
    const float* W        = (const float*)d_in[4];
    const float* b        = (const float*)d_in[5];
    const float* bn_gamma = (const float*)d_in[6];
    const float* bn_beta  = (const float*)d_in[7];
    const float* ph_W1    = (const float*)d_in[8];
    const float* ph_b1    = (const float*)d_in[9];
    const float* ph_W2    = (const float*)d_in[10];
    const float* ph_b2    = (const float*)d_in[11];

    const int N = in_sizes[0] / H;
    const int E = in_sizes[2];
    const int C = H * (LAYERS + 1);           // 512

    const int* row = eidx;
    const int* col = eidx + E;

    // Workspace carve-up
    float* dinv  = (float*)d_ws;              // N  (degree, then 1/sqrt(deg))
    float* selfn = dinv + N;                  // N
    float* normw = selfn + N;                 // E
    float* zbuf  = normw + E;                 // N*H
    float* hbuf  = zbuf + (size_t)N * H;      // N*H
    float* aggb  = hbuf + (size_t)N * H;      // N*H
    float* stat  = aggb + (size_t)N * H;      // 2*H

    float* zcat = (float*)d_out;              // [N, 512]
    float* gcat = zcat + (size_t)N * C;       // [G, 512]

    const int T = 256;
    const int gN   = (N + T - 1) / T;
    const int gE   = (E + T - 1) / T;
    const long long NH = (long long)N * H;
    const int gNH  = (int)((NH + T - 1) / T);
    const int gE32 = (int)(((long long)E * 32 + T - 1) / T);
    const int gGEMM = (N + 15) / 16;

    // ---- normalization precompute ----
    fill_one_kernel<<<gN, T, 0, stream>>>(dinv, N);
    deg_accum_kernel<<<gE, T, 0, stream>>>(col, eattr, dinv, E);
    deg_finalize_kernel<<<gN, T, 0, stream>>>(dinv, selfn, N);
    edge_norm_kernel<<<gE, T, 0, stream>>>(row, col, eattr, dinv, normw, E);

    // ---- GCN layers ----
    const float* zin = x;
    for (int i = 0; i < LAYERS; ++i) {
        gemm_wmma_kernel<<<gGEMM, T, 0, stream>>>(
            zin, W + (size_t)i * H * H, nullptr, hbuf, N, H, 0, 0);
        agg_init_kernel<<<gNH, T, 0, stream>>>(hbuf, selfn, b + (size_t)i * H, aggb, N);
        scatter_kernel<<<gE32, T, 0, stream>>>(row, col, normw, hbuf, aggb, E);
        zero_kernel<<<1, 256, 0, stream>>>(stat, 2 * H);
        relu_stats_kernel<<<(N + 63) / 64, T, 0, stream>>>(aggb, stat, N);
        bn_finalize_kernel<<<1, H, 0, stream>>>(
            stat, bn_gamma + (size_t)i * H, bn_beta + (size_t)i * H, N);
        bn_apply_kernel<<<gNH, T, 0, stream>>>(aggb, stat, zbuf, zcat, i, N);
        zin = zbuf;
    }

    // ---- projection head: relu(z@W1+b1) @ W2 + b2, written into z_cat[:,384:512]
    gemm_wmma_kernel<<<gGEMM, T, 0, stream>>>(zbuf, ph_W1, ph_b1, hbuf, N, H, 0, 1);
    gemm_wmma_kernel<<<gGEMM, T, 0, stream>>>(hbuf, ph_W2, ph_b2, zcat, N, C, LAYERS * H, 0);

    // ---- graph pooling ----
    long long gElems = (long long)GRAPHS * C;
    zero_kernel<<<(int)((gElems + T - 1) / T), T, 0, stream>>>(gcat, gElems);
    pool_kernel<<<(int)((NH + T - 1) / T), T, 0, stream>>>(zcat, batch, gcat, N);
}